// AdaptiveCornerLoss_24919400252186
// MI455X (gfx1250) — compile-verified
//
#include <hip/hip_runtime.h>
#include <hip/hip_bf16.h>
#include <math.h>

typedef __attribute__((ext_vector_type(2))) float v2f;
typedef __attribute__((ext_vector_type(8))) float v8f;

#define WAVES_PER_BLOCK 8
#define GRID_Y 16

// Problem constants (from reference setup_inputs)
#define BB 64
#define NN 16384
#define CC 6
#define MM 128

__global__ void acl_init_kernel(float* acc) {
    acc[0] = 0.0f;
    acc[1] = 0.0f;
}

__global__ __launch_bounds__(256) void acl_main_kernel(
        const float* __restrict__ inputs,
        const float* __restrict__ targets,
        const float* __restrict__ pcoords,
        const float* __restrict__ ccoords,
        float* __restrict__ acc)
{
    const int b    = blockIdx.x;                 // batch index
    const int lane = threadIdx.x & 31;
    const int wv   = threadIdx.x >> 5;
    const int n16  = lane & 15;
    const bool hi  = lane >= 16;

    // two 16x16 min-tiles per wave, padded to dodge bank conflicts
    __shared__ float red[WAVES_PER_BLOCK][2][16][17];

    // ---- Build WMMA B-matrix operands for all 128 corners (8 tiles of 16) ----
    // B is 4x16 (KxN), f32, 2 VGPRs: VGPR0 = {K=0 | K=2}, VGPR1 = {K=1 | K=3},
    // column n = lane&15.  B column j = [-2c0, -2c1, -2c2, csq].
    v2f Bm[8];
    const float* cb = ccoords + (size_t)b * MM * 3;
    #pragma unroll
    for (int t = 0; t < 8; ++t) {
        const int j = t * 16 + n16;
        const float c0 = cb[j * 3 + 0];
        const float c1 = cb[j * 3 + 1];
        const float c2 = cb[j * 3 + 2];
        float csq = c0 * c0 + c1 * c1 + c2 * c2;
        csq = (c0 > -1.0f) ? csq : 1e30f;        // invalid corner -> never wins min
        Bm[t].x = hi ? (-2.0f * c2) : (-2.0f * c0);
        Bm[t].y = hi ? csq          : (-2.0f * c1);
    }

    float sF = 0.0f, sD = 0.0f;
    const int   waveGlobal = blockIdx.y * WAVES_PER_BLOCK + wv;
    const int   waveStride = gridDim.y * WAVES_PER_BLOCK;
    const float* pb = pcoords + (size_t)b * NN * CC;
    const float* ib = inputs  + (size_t)b * NN;
    const float* tb = targets + (size_t)b * NN;

    // 32 points (two 16-row WMMA tiles) per iteration -> all lanes busy in the tail
    for (int pt = waveGlobal; pt < NN / 32; pt += waveStride) {
        const int base = pt * 32;
        const int i0   = base + n16;        // tile 0 row for this lane
        const int i1   = base + 16 + n16;   // tile 1 row for this lane

        // speculative prefetch of the next iteration's coords (global_prefetch_b8)
        __builtin_prefetch(pb + (size_t)(base + waveStride * 32) * CC, 0, 1);

        // ---- A-matrices: 16x4 (MxK) f32, 2 VGPRs: VGPR0 = {K=0 | K=2}, VGPR1 = {K=1 | K=3}
        // row m = lane&15 (both half-waves load the same point). A row = [p0,p1,p2,1].
        const float q0 = pb[(size_t)i0 * CC + 0];
        const float q1 = pb[(size_t)i0 * CC + 1];
        const float q2 = pb[(size_t)i0 * CC + 2];
        const float r0 = pb[(size_t)i1 * CC + 0];
        const float r1 = pb[(size_t)i1 * CC + 1];
        const float r2 = pb[(size_t)i1 * CC + 2];
        v2f Am0, Am1;
        Am0.x = hi ? q2   : q0;
        Am0.y = hi ? 1.0f : q1;
        Am1.x = hi ? r2   : r0;
        Am1.y = hi ? 1.0f : r1;
        const float psq0 = q0 * q0 + q1 * q1 + q2 * q2;
        const float psq1 = r0 * r0 + r1 * r1 + r2 * r2;

        // ---- 16x V_WMMA_F32_16X16X4_F32: D[m][n] = csq[n] - 2*p[m]·c[n], min over corner tiles
        v8f czero = {};
        v8f mn0 = __builtin_amdgcn_wmma_f32_16x16x4_f32(
            false, Am0, false, Bm[0], (short)0, czero, false, false);
        v8f mn1 = __builtin_amdgcn_wmma_f32_16x16x4_f32(
            false, Am1, false, Bm[0], (short)0, czero, false, false);
        #pragma unroll
        for (int t = 1; t < 8; ++t) {
            v8f d0 = __builtin_amdgcn_wmma_f32_16x16x4_f32(
                false, Am0, false, Bm[t], (short)0, czero, false, false);
            v8f d1 = __builtin_amdgcn_wmma_f32_16x16x4_f32(
                false, Am1, false, Bm[t], (short)0, czero, false, false);
            #pragma unroll
            for (int v = 0; v < 8; ++v) {
                mn0[v] = fminf(mn0[v], d0[v]);
                mn1[v] = fminf(mn1[v], d1[v]);
            }
        }

        // ---- Transpose through LDS: lane holds D rows (v + 8*hi), column n16.
        const int row0 = hi ? 8 : 0;
        #pragma unroll
        for (int v = 0; v < 8; ++v) {
            red[wv][0][row0 + v][n16] = mn0[v];
            red[wv][1][row0 + v][n16] = mn1[v];
        }

        // DS ops are in-order per wave; wait for stores to land + compiler fence.
        asm volatile("s_wait_dscnt 0" ::: "memory");

        // ---- every lane owns one point: lanes 0..15 -> tile0 rows, 16..31 -> tile1 rows
        const int sel   = hi ? 1 : 0;
        const int i     = base + lane;        // == base + 16*sel + n16
        const float psq = hi ? psq1 : psq0;

        float m = red[wv][sel][n16][0];
        #pragma unroll
        for (int c = 1; c < 16; ++c) m = fminf(m, red[wv][sel][n16][c]);
        const float d2   = fmaxf(m + psq, 1e-12f);
        const float mind = sqrtf(d2);
        const float w    = 1.0f + 2.0f * __expf(-mind * 10.0f);  // 1/DIST_THRESHOLD = 10

        const float x  = ib[i];
        const float tg = tb[i];
        const float ce = fmaxf(x, 0.0f) - x * tg + log1pf(__expf(-fabsf(x)));
        const float pr = 1.0f / (1.0f + __expf(-x));
        const float pt2 = pr * tg + (1.0f - pr) * (1.0f - tg);
        const float om = 1.0f - pt2;
        const float fe = om * om * ce;       // ALPHA=1, GAMMA=2
        sF += fe;
        sD += fe * w;

        // keep next iteration's LDS stores from drifting above this tile's reads
        __builtin_amdgcn_wave_barrier();
    }

    // ---- wave32 reduction, one atomic pair per wave
    #pragma unroll
    for (int off = 16; off > 0; off >>= 1) {
        sF += __shfl_xor(sF, off, 32);
        sD += __shfl_xor(sD, off, 32);
    }
    if (lane == 0) {
        atomicAdd(&acc[0], sF);
        atomicAdd(&acc[1], sD);
    }
}

__global__ void acl_final_kernel(const float* __restrict__ acc, float* __restrict__ out) {
    const float inv   = 1.0f / (float)((long long)BB * NN);
    const float focal = acc[0] * inv;
    const float dist  = acc[1] * inv;
    out[0] = focal + 1.0f * dist;   // total (DISTANCE_WEIGHT = 1)
    out[1] = focal;
    out[2] = dist;
}

extern "C" void kernel_launch(void* const* d_in, const int* in_sizes, int n_in,
                              void* d_out, int out_size, void* d_ws, size_t ws_size,
                              hipStream_t stream) {
    const float* inputs  = (const float*)d_in[0];
    const float* targets = (const float*)d_in[1];
    const float* pcoords = (const float*)d_in[2];
    const float* ccoords = (const float*)d_in[3];
    float* out = (float*)d_out;
    float* acc = (float*)d_ws;

    acl_init_kernel<<<1, 1, 0, stream>>>(acc);
    dim3 grid(BB, GRID_Y);
    acl_main_kernel<<<grid, 32 * WAVES_PER_BLOCK, 0, stream>>>(
        inputs, targets, pcoords, ccoords, acc);
    acl_final_kernel<<<1, 1, 0, stream>>>(acc, out);
}